// AdaptiveCurvatureLoss_52733608460557
// MI455X (gfx1250) — compile-verified
//
#include <hip/hip_runtime.h>

typedef __attribute__((ext_vector_type(2))) float v2f;
typedef __attribute__((ext_vector_type(8))) float v8f;

#define EPSF 1e-8f

// Branch-free tanh: tanh(u) = 1 - 2/(1 + e^{2u}).
// exp2 overflow -> +inf -> rcp -> 0 -> t = +1 ; underflow -> 0 -> t = -1.
// Two transcendental VALU ops (v_exp_f32, v_rcp_f32), no EXEC divergence.
__device__ __forceinline__ float fast_tanhf(float u) {
    const float LOG2E_X2 = 2.885390081777926815f;  // 2*log2(e)
    float e = __builtin_amdgcn_exp2f(u * LOG2E_X2);
    float r = __builtin_amdgcn_rcpf(1.0f + e);
    return 1.0f - 2.0f * r;
}

// ---------------------------------------------------------------------------
// Kernel 1: fused MLP forward + analytic 2nd derivative.
// Each 32-lane wave owns a tile of 16 x-values. Per K-chunk of 4 hidden units,
// each lane computes two tanh activations (A-matrix 16x4 f32 layout: lanes
// 0-15 hold K=0,1; lanes 16-31 hold K=2,3 for M=lane&15), folds the output
// weights into A, and accumulates a row-sum with V_WMMA_F32_16X16X4_F32
// against an all-ones B matrix:
//   y_M  = sum_h w2_h * tanh(w1_h x_M + b1_h)
//   d2_M = sum_h (w2_h w1_h^2) * (-2 t (1 - t^2))
// Per-block partial sums of (y - t)^2 and d2^2 go to workspace.
// ---------------------------------------------------------------------------
__global__ __launch_bounds__(128) void mlp_d2_kernel(
    const float* __restrict__ x, const float* __restrict__ targets,
    const float* __restrict__ w1, const float* __restrict__ b1,
    const float* __restrict__ w2, const float* __restrict__ b2,
    float* __restrict__ part_e2, float* __restrict__ part_d22)
{
    const int tid  = threadIdx.x;
    const int wave = tid >> 5;
    const int lane = tid & 31;
    const int half = lane >> 4;     // which half of the wave
    const int mlo  = lane & 15;     // M row this lane feeds in the A matrix
    const int i0   = blockIdx.x * 64 + wave * 16;

    const float xM = x[i0 + mlo];

    v8f accY, accD;
#pragma unroll
    for (int r = 0; r < 8; ++r) { accY[r] = 0.0f; accD[r] = 0.0f; }
    const v2f ones = {1.0f, 1.0f};

    const int kb = half * 2;        // K offset inside each 4-wide chunk
#pragma unroll
    for (int c = 0; c < 16; ++c) {  // 16 chunks x K=4 -> H=64
        const int h0 = 4 * c + kb;
        const int h1 = h0 + 1;
        const float W10 = w1[h0], W11 = w1[h1];
        const float B10 = b1[h0], B11 = b1[h1];
        const float W20 = w2[h0], W21 = w2[h1];
        const float t0 = fast_tanhf(fmaf(W10, xM, B10));
        const float t1 = fast_tanhf(fmaf(W11, xM, B11));
        v2f aY = { W20 * t0, W21 * t1 };
        const float s0 = 1.0f - t0 * t0;
        const float s1 = 1.0f - t1 * t1;
        v2f aD = { (W20 * W10 * W10) * (-2.0f * t0 * s0),
                   (W21 * W11 * W11) * (-2.0f * t1 * s1) };
        accY = __builtin_amdgcn_wmma_f32_16x16x4_f32(
            false, aY, false, ones, (short)0, accY, false, false);
        accD = __builtin_amdgcn_wmma_f32_16x16x4_f32(
            false, aD, false, ones, (short)0, accD, false, false);
    }

    // D layout: lane half h holds M = r + 8*h in VGPR r (all 16 lanes of a
    // half duplicate the same values -> scale partial sums by 1/16).
    const float b2v = b2[0];
    float pe2 = 0.0f, pd2 = 0.0f;
#pragma unroll
    for (int r = 0; r < 8; ++r) {
        const int i = i0 + half * 8 + r;
        const float e = accY[r] + b2v - targets[i];
        pe2 += e * e;
        const float d2 = accD[r];
        pd2 += d2 * d2;
    }
    pe2 *= (1.0f / 16.0f);
    pd2 *= (1.0f / 16.0f);

    __shared__ float rE[128];
    __shared__ float rD[128];
    rE[tid] = pe2; rD[tid] = pd2;
    __syncthreads();
    for (int s = 64; s > 0; s >>= 1) {
        if (tid < s) { rE[tid] += rE[tid + s]; rD[tid] += rD[tid + s]; }
        __syncthreads();
    }
    if (tid == 0) {
        part_e2[blockIdx.x]  = rE[0];
        part_d22[blockIdx.x] = rD[0];
    }
}

// ---------------------------------------------------------------------------
// Kernel 2: streaming top-3 nearest |x_i - x_j| (includes self-distance EPS,
// matching the reference's diagonal). x tiles through LDS (gfx1250 async
// global->LDS loads when available); no NxN allocation. Branch-free top-3
// insertion network. Emits per-block sum(dens) and max(dens).
// ---------------------------------------------------------------------------
#if defined(__gfx1250__) &&                                            \
    __has_builtin(__builtin_amdgcn_global_load_async_to_lds_b128) &&   \
    __has_builtin(__builtin_amdgcn_s_wait_asynccnt)
#define USE_ASYNC_LDS 1
typedef int v4i __attribute__((ext_vector_type(4)));
// Pointee-qualified pointers: AS(1) = global ("__device__"), AS(3) = LDS.
typedef __attribute__((address_space(1))) v4i* gv4ip;
typedef __attribute__((address_space(3))) v4i* lv4ip;
#else
#define USE_ASYNC_LDS 0
#endif

__global__ __launch_bounds__(256) void density_kernel(
    const float* __restrict__ x, int n,
    float* __restrict__ part_sum, float* __restrict__ part_max)
{
    __shared__ __attribute__((aligned(16))) float xs[1024];
    __shared__ float rS[256];
    __shared__ float rM[256];
    const int tid = threadIdx.x;
    const int i   = blockIdx.x * 256 + tid;
    const float xi = x[i];

    float a = 3.0e38f, b = 3.0e38f, c = 3.0e38f;   // running top-3 ascending
    for (int t = 0; t < n; t += 1024) {
#if USE_ASYNC_LDS
        // Per-lane 16B global->LDS DMA (GLOBAL_LOAD_ASYNC_TO_LDS_B128),
        // tracked by ASYNCcnt, drained with s_wait_asynccnt 0.
        __builtin_amdgcn_global_load_async_to_lds_b128(
            (gv4ip)(x + t + (tid << 2)), (lv4ip)&xs[tid << 2], 0, 0);
        __builtin_amdgcn_s_wait_asynccnt(0);
#else
#pragma unroll
        for (int k = 0; k < 4; ++k)
            xs[tid + 256 * k] = x[t + tid + 256 * k];
#endif
        __syncthreads();
#pragma unroll 8
        for (int j = 0; j < 1024; ++j) {
            const float d = fabsf(xi - xs[j]) + EPSF;
            // branchless insert of d into ascending (a,b,c), keep 3 smallest
            const float m1 = fmaxf(a, d);  a = fminf(a, d);
            const float m2 = fmaxf(b, m1); b = fminf(b, m1);
            c = fminf(c, m2);
        }
        __syncthreads();
    }

    const float m    = (a + b + c) * (1.0f / 3.0f);
    const float dens = 1.0f / (m + EPSF);

    rS[tid] = dens; rM[tid] = dens;
    __syncthreads();
    for (int s = 128; s > 0; s >>= 1) {
        if (tid < s) {
            rS[tid] += rS[tid + s];
            rM[tid]  = fmaxf(rM[tid], rM[tid + s]);
        }
        __syncthreads();
    }
    if (tid == 0) {
        part_sum[blockIdx.x] = rS[0];
        part_max[blockIdx.x] = rM[0];
    }
}

// ---------------------------------------------------------------------------
// Kernel 3: deterministic fixed-order reduction of all partials + final math.
// mean(adaptive_w) = 1 + 0.1 * (mean(dens) / (max(dens)+EPS));
// penalty = 0.01 * mean(adaptive_w) * mean(d2^2); total = mse + penalty.
// ---------------------------------------------------------------------------
__global__ __launch_bounds__(256) void finalize_kernel(
    const float* __restrict__ ws, int nb1, int nb2, int n,
    float* __restrict__ out)
{
    __shared__ float rA[256], rB[256], rC[256], rD[256];
    const int t = threadIdx.x;
    const float* e2  = ws;
    const float* d22 = ws + nb1;
    const float* dsS = ws + 2 * nb1;
    const float* dsM = ws + 2 * nb1 + nb2;

    float sE = 0.0f, sD = 0.0f;
    for (int k = t; k < nb1; k += 256) { sE += e2[k]; sD += d22[k]; }
    float sS = 0.0f, mx = 0.0f;
    for (int k = t; k < nb2; k += 256) { sS += dsS[k]; mx = fmaxf(mx, dsM[k]); }

    rA[t] = sE; rB[t] = sD; rC[t] = sS; rD[t] = mx;
    __syncthreads();
    for (int s = 128; s > 0; s >>= 1) {
        if (t < s) {
            rA[t] += rA[t + s];
            rB[t] += rB[t + s];
            rC[t] += rC[t + s];
            rD[t]  = fmaxf(rD[t], rD[t + s]);
        }
        __syncthreads();
    }
    if (t == 0) {
        const float invN     = 1.0f / (float)n;
        const float mse      = rA[0] * invN;
        const float meanD2sq = rB[0] * invN;
        const float densMean = rC[0] * invN;
        const float densMax  = rD[0];
        const float meanW    = 1.0f + 0.1f * (densMean / (densMax + EPSF));
        const float pen      = 0.01f * meanW * meanD2sq;
        out[0] = mse + pen;
        out[1] = mse;
        out[2] = pen;
    }
}

extern "C" void kernel_launch(void* const* d_in, const int* in_sizes, int n_in,
                              void* d_out, int out_size, void* d_ws, size_t ws_size,
                              hipStream_t stream) {
    const float* x  = (const float*)d_in[0];
    const float* tg = (const float*)d_in[1];
    const float* w1 = (const float*)d_in[2];
    const float* b1 = (const float*)d_in[3];
    const float* w2 = (const float*)d_in[4];
    const float* b2 = (const float*)d_in[5];
    float* ws  = (float*)d_ws;
    float* out = (float*)d_out;

    const int n   = in_sizes[0];   // 16384
    const int nb1 = n / 64;        // blocks for kernel 1 (64 elems/block)
    const int nb2 = n / 256;       // blocks for kernel 2 (256 elems/block)

    // ws layout (floats): [0,nb1) e2 partials | [nb1,2nb1) d2^2 partials |
    //                     [2nb1,2nb1+nb2) dens sums | [.., +nb2) dens maxes
    mlp_d2_kernel<<<nb1, 128, 0, stream>>>(x, tg, w1, b1, w2, b2, ws, ws + nb1);
    density_kernel<<<nb2, 256, 0, stream>>>(x, n, ws + 2 * nb1, ws + 2 * nb1 + nb2);
    finalize_kernel<<<1, 256, 0, stream>>>(ws, nb1, nb2, n, out);
}